// RecPolicy_20761871909590
// MI455X (gfx1250) — compile-verified
//
#include <hip/hip_runtime.h>
#include <hip/hip_bf16.h>

typedef float v2f __attribute__((ext_vector_type(2)));
typedef float v8f __attribute__((ext_vector_type(8)));

// D = A(16x4 f32) * B(4x16 f32) + C(16x16 f32), fp32 accumulate, RNE.
#define WMMA4(a, b, c) \
  __builtin_amdgcn_wmma_f32_16x16x4_f32(false, (a), false, (b), (short)0, (c), false, false)

__global__ __launch_bounds__(32) void recpolicy_kernel(
    const float* __restrict__ x,
    const float* __restrict__ upWih, const float* __restrict__ upWhh,
    const float* __restrict__ upBih, const float* __restrict__ upBhh,
    const float* __restrict__ f1W,   const float* __restrict__ f1b,
    const float* __restrict__ f2W,   const float* __restrict__ f2b,
    const float* __restrict__ dnWih, const float* __restrict__ dnWhh,
    const float* __restrict__ dnBih, const float* __restrict__ dnBhh,
    const float* __restrict__ unW,   const float* __restrict__ unb,
    float* __restrict__ out)
{
  const int lane = threadIdx.x;          // 0..31, single wave32
  const int row  = lane & 15;            // A-matrix M index
  const int kb   = (lane >> 4) << 1;     // half-wave K pair base: 0 or 2

  __shared__ float sX[64];               // staged input vector x (57)
  __shared__ float sV[24];               // current matvec RHS (padded)
  __shared__ float sH[8];                // current hidden state
  __shared__ float sHup[4][8];           // up-pass outputs h_up[0..3]
  __shared__ float sPre[16];             // down-pass fused pre-activations
  __shared__ float sBup[8], sBdn[8], sB1[8], sB2[8], sUnb[8];

  // ---- stage x and biases into LDS ----
  for (int i = lane; i < 57; i += 32) sX[i] = x[i];
  if (lane < 8) {
    sBup[lane] = upBih[lane] + upBhh[lane];
    sBdn[lane] = dnBih[lane] + dnBhh[lane];
    sB1[lane]  = f1b[lane];
    sB2[lane]  = f2b[lane];
    sH[lane]   = 0.0f;                   // up-scan initial carry
    sHup[3][lane] = 0.0f;                // reference: h_up[3] == zeros
  }
  if (lane < 6) sUnb[lane] = unb[lane];

  // ---- preload A operands in 16x4 f32 WMMA layout (row = lane&15, K pair by half-wave) ----
  // UP fused matrix 8x20: cols 0..11 = r_up_Wih, cols 12..19 = r_up_Whh
  v2f aUp[5];
#pragma unroll
  for (int kc = 0; kc < 5; ++kc) {
    int k0 = kc * 4 + kb, k1 = k0 + 1;
    float a0 = 0.0f, a1 = 0.0f;
    if (row < 8) {
      a0 = (k0 < 12) ? upWih[row * 12 + k0] : upWhh[row * 8 + (k0 - 12)];
      a1 = (k1 < 12) ? upWih[row * 12 + k1] : upWhh[row * 8 + (k1 - 12)];
    }
    aUp[kc] = (v2f){a0, a1};
  }
  // FC1 8x13 (K padded to 16)
  v2f aF1[4];
#pragma unroll
  for (int kc = 0; kc < 4; ++kc) {
    int k0 = kc * 4 + kb, k1 = k0 + 1;
    float a0 = (row < 8 && k0 < 13) ? f1W[row * 13 + k0] : 0.0f;
    float a1 = (row < 8 && k1 < 13) ? f1W[row * 13 + k1] : 0.0f;
    aF1[kc] = (v2f){a0, a1};
  }
  // FC2 8x8
  v2f aF2[2];
#pragma unroll
  for (int kc = 0; kc < 2; ++kc) {
    int k0 = kc * 4 + kb, k1 = k0 + 1;
    float a0 = (row < 8) ? f2W[row * 8 + k0] : 0.0f;
    float a1 = (row < 8) ? f2W[row * 8 + k1] : 0.0f;
    aF2[kc] = (v2f){a0, a1};
  }
  // DOWN fused matrix 14x16: rows 0..5 = un_W (acts on concat(h,hu));
  // rows 6..13: cols 0..7 = r_down_Whh (h), cols 8..15 = r_down_Wih (hu)
  v2f aDn[4];
#pragma unroll
  for (int kc = 0; kc < 4; ++kc) {
    int k0 = kc * 4 + kb, k1 = k0 + 1;
    float a0 = 0.0f, a1 = 0.0f;
    if (row < 6) {
      a0 = unW[row * 16 + k0];
      a1 = unW[row * 16 + k1];
    } else if (row < 14) {
      int r = row - 6;
      a0 = (k0 < 8) ? dnWhh[r * 8 + k0] : dnWih[r * 8 + (k0 - 8)];
      a1 = (k1 < 8) ? dnWhh[r * 8 + k1] : dnWih[r * 8 + (k1 - 8)];
    }
    aDn[kc] = (v2f){a0, a1};
  }
  __syncthreads();

  // ---- up pass (reference scan in reverse): h_up[t] = cell(local[t+1], h_up[t+1]) ----
  for (int t = 2; t >= 0; --t) {
    // v[0..11] = local[t+1] (jl | jdl with the 2-zero front shift), v[12..19] = h
    if (lane < 12) {
      float val;
      if (lane < 6) { int f = (t + 1) * 6 + lane;       val = (f < 2) ? 0.0f : sX[f + 5];  }
      else          { int f = (t + 1) * 6 + (lane - 6); val = (f < 2) ? 0.0f : sX[f + 33]; }
      sV[lane] = val;
    } else if (lane < 20) {
      sV[lane] = sH[lane - 12];
    }
    __syncthreads();

    v8f acc = {};
#pragma unroll
    for (int kc = 0; kc < 5; ++kc) {
      v2f b = (v2f){sV[kc * 4 + kb], sV[kc * 4 + kb + 1]};   // columns all equal
      acc = WMMA4(aUp[kc], b, acc);
    }
    if (lane == 0) {
#pragma unroll
      for (int j = 0; j < 8; ++j) {
        float hn = tanhf(acc[j] + sBup[j]);
        sH[j] = hn;
        sHup[t][j] = hn;
      }
    }
    __syncthreads();
  }

  // ---- MLP: h0 = tanh(fc2 @ tanh(fc1 @ obs_cat + b1) + b2) ----
  if (lane < 16) {
    float val = 0.0f;
    if (lane < 7)       val = sX[lane];            // obs  = x[0:7]
    else if (lane < 13) val = sX[29 + (lane - 7)]; // obsd = x[29:35]
    sV[lane] = val;
  }
  __syncthreads();
  {
    v8f acc = {};
#pragma unroll
    for (int kc = 0; kc < 4; ++kc) {
      v2f b = (v2f){sV[kc * 4 + kb], sV[kc * 4 + kb + 1]};
      acc = WMMA4(aF1[kc], b, acc);
    }
    __syncthreads();
    if (lane == 0) {
#pragma unroll
      for (int j = 0; j < 8; ++j) sV[j] = tanhf(acc[j] + sB1[j]);
    }
    __syncthreads();
    v8f acc2 = {};
#pragma unroll
    for (int kc = 0; kc < 2; ++kc) {
      v2f b = (v2f){sV[kc * 4 + kb], sV[kc * 4 + kb + 1]};
      acc2 = WMMA4(aF2[kc], b, acc2);
    }
    if (lane == 0) {
#pragma unroll
      for (int j = 0; j < 8; ++j) sH[j] = tanhf(acc2[j] + sB2[j]);
    }
    __syncthreads();
  }

  // ---- down pass: one fused 14x16 WMMA chain per step -> act + h_new pre-activation ----
  for (int i = 0; i < 4; ++i) {
    if (lane < 8)       sV[lane] = sH[lane];            // h
    else if (lane < 16) sV[lane] = sHup[i][lane - 8];   // hu
    __syncthreads();

    v8f acc = {};
#pragma unroll
    for (int kc = 0; kc < 4; ++kc) {
      v2f b = (v2f){sV[kc * 4 + kb], sV[kc * 4 + kb + 1]};
      acc = WMMA4(aDn[kc], b, acc);
    }
    // D rows 0..7 live in lanes 0-15 (all identical), rows 8..13 in lanes 16-31
    if (lane == 0) {
#pragma unroll
      for (int j = 0; j < 8; ++j) sPre[j] = acc[j];
    }
    if (lane == 16) {
#pragma unroll
      for (int j = 0; j < 6; ++j) sPre[8 + j] = acc[j];
    }
    __syncthreads();

    if (lane < 6) {                       // acts.flat[2:] -> out
      int oi = i * 6 + lane - 2;
      if (oi >= 0) out[oi] = sPre[lane] + sUnb[lane];
    } else if (lane < 14) {               // h_new = tanh(Wih@hu + Whh@h + b)
      int j = lane - 6;
      sH[j] = tanhf(sPre[6 + j] + sBdn[j]);
    }
    __syncthreads();
  }
}

extern "C" void kernel_launch(void* const* d_in, const int* in_sizes, int n_in,
                              void* d_out, int out_size, void* d_ws, size_t ws_size,
                              hipStream_t stream) {
  (void)in_sizes; (void)n_in; (void)out_size; (void)d_ws; (void)ws_size;
  recpolicy_kernel<<<1, 32, 0, stream>>>(
      (const float*)d_in[0],
      (const float*)d_in[1],  (const float*)d_in[2],
      (const float*)d_in[3],  (const float*)d_in[4],
      (const float*)d_in[5],  (const float*)d_in[6],
      (const float*)d_in[7],  (const float*)d_in[8],
      (const float*)d_in[9],  (const float*)d_in[10],
      (const float*)d_in[11], (const float*)d_in[12],
      (const float*)d_in[13], (const float*)d_in[14],
      (float*)d_out);
}